// GCN_61177514164633
// MI455X (gfx1250) — compile-verified
//
#include <hip/hip_runtime.h>
#include <math.h>

// ---------------------------------------------------------------------------
// GCN forward for MI455X (gfx1250, wave32).
// Dense GEMMs use V_WMMA_F32_16X16X4_F32 (full fp32 precision; node matrices
// fit in the 192MB L2 so the message-passing atomics dominate, not the GEMM).
// ---------------------------------------------------------------------------

typedef float v2f __attribute__((ext_vector_type(2)));
typedef float v8f __attribute__((ext_vector_type(8)));

// ---------------- degree / normalization ----------------

__global__ void deg_init_kernel(float* deg, int n) {
  int i = blockIdx.x * blockDim.x + threadIdx.x;
  if (i < n) deg[i] = 1.0f;  // self-loop contributes 1 to every node
}

__global__ void deg_edge_kernel(const int* __restrict__ dst, float* deg, int E) {
  int i = blockIdx.x * blockDim.x + threadIdx.x;
  if (i < E) atomicAdd(&deg[dst[i]], 1.0f);
}

__global__ void dinv_kernel(const float* __restrict__ deg, float* dinv, int n) {
  int i = blockIdx.x * blockDim.x + threadIdx.x;
  if (i < n) {
    float d = deg[i];
    dinv[i] = d > 0.0f ? rsqrtf(d) : 0.0f;
  }
}

// ---------------- dense GEMM via WMMA f32 16x16x4 ----------------
// C[M,N] = A[M,K] @ B[K,N].  One wave computes one 16x16 output tile.
// Block = (N/16) waves covering a 16-row strip of C; grid.x = M/16.
// Fragment layouts per CDNA5 ISA 7.12.2:
//   A (16x4, f32):  M = lane&15, VGPR{0,1} hold K = 2*(lane>>4) + {0,1}
//   B (4x16, f32):  N = lane&15, VGPR{0,1} hold K = 2*(lane>>4) + {0,1}
//   C/D (16x16):    N = lane&15, VGPR r holds M = r + 8*(lane>>4)
__global__ void gemm_wmma_f32_kernel(const float* __restrict__ A,
                                     const float* __restrict__ B,
                                     float* __restrict__ C,
                                     int M, int K, int N) {
  const int lane = threadIdx.x & 31;
  const int wave = threadIdx.x >> 5;
  const int ntiles = N >> 4;
  if (wave >= ntiles) return;           // whole-wave exit: EXEC stays all-ones
  const int mbase = blockIdx.x << 4;
  const int nbase = wave << 4;
  const int mrow  = mbase + (lane & 15);
  const int ncol  = nbase + (lane & 15);
  const int khalf = (lane >> 4) << 1;   // 0 for lanes 0-15, 2 for lanes 16-31

  v8f acc = {0.f, 0.f, 0.f, 0.f, 0.f, 0.f, 0.f, 0.f};
  const float* Arow = A + (size_t)mrow * K;
  for (int k = 0; k < K; k += 4) {
    v2f a = *(const v2f*)(Arow + k + khalf);     // {A[m][k+kh], A[m][k+kh+1]}
    v2f b;
    b.x = B[(size_t)(k + khalf) * N + ncol];
    b.y = B[(size_t)(k + khalf + 1) * N + ncol];
    acc = __builtin_amdgcn_wmma_f32_16x16x4_f32(
        /*neg_a=*/false, a, /*neg_b=*/false, b,
        /*c_mod=*/(short)0, acc, /*reuse_a=*/false, /*reuse_b=*/false);
  }

  const int row0 = mbase + ((lane >> 4) << 3);   // +8 rows for upper half-wave
#pragma unroll
  for (int r = 0; r < 8; ++r)
    C[(size_t)(row0 + r) * N + ncol] = acc[r];
}

// ---------------- edge scatter:  out[dst] += hw[src] * dinv[src]*dinv[dst] ---
// One WAVE per edge: edge id is wave-uniform (scalarizable src/dst loads,
// no per-thread div/mod), 32 lanes x float4 cover the 128-float row with a
// coalesced 512B read and row-contiguous L2 atomics.
__global__ void scatter_edge_kernel(const int* __restrict__ src,
                                    const int* __restrict__ dst,
                                    const float* __restrict__ dinv,
                                    const float* __restrict__ hw,
                                    float* __restrict__ out,
                                    int E, int F) {
  const int wavesPerBlock = blockDim.x >> 5;
  const int e = blockIdx.x * wavesPerBlock + (threadIdx.x >> 5);
  if (e >= E) return;
  const int lane = threadIdx.x & 31;            // lane == float4 chunk (F==128)
  const int s = src[e];
  const int d = dst[e];
  const float nrm = dinv[s] * dinv[d];
  const float4 v = *(const float4*)(hw + (size_t)s * F + (lane << 2));
  float* o = out + (size_t)d * F + (lane << 2);
  atomicAdd(o + 0, v.x * nrm);
  atomicAdd(o + 1, v.y * nrm);
  atomicAdd(o + 2, v.z * nrm);
  atomicAdd(o + 3, v.w * nrm);
}

// -------- epilogue: add self-loop (dinv[i]^2 * hw[i]), bias, optional ReLU --
// One wave per node row; float4 (b128) per lane.
__global__ void bias_self_act_kernel(float* __restrict__ out,
                                     const float* __restrict__ hw,
                                     const float* __restrict__ dinv,
                                     const float* __restrict__ bias,
                                     int n_nodes, int F, int do_relu) {
  const int wavesPerBlock = blockDim.x >> 5;
  const int node = blockIdx.x * wavesPerBlock + (threadIdx.x >> 5);
  if (node >= n_nodes) return;
  const int lane = threadIdx.x & 31;
  const float di = dinv[node];
  const float dd = di * di;
  const size_t base = (size_t)node * F + (lane << 2);
  float4 o = *(float4*)(out + base);
  const float4 w = *(const float4*)(hw + base);
  const float4 b = *(const float4*)(bias + (lane << 2));
  o.x += w.x * dd + b.x;
  o.y += w.y * dd + b.y;
  o.z += w.z * dd + b.z;
  o.w += w.w * dd + b.w;
  if (do_relu) {
    o.x = fmaxf(o.x, 0.0f);
    o.y = fmaxf(o.y, 0.0f);
    o.z = fmaxf(o.z, 0.0f);
    o.w = fmaxf(o.w, 0.0f);
  }
  *(float4*)(out + base) = o;
}

// ---------------- global mean pool via sorted-run strips ----------------
// batch is sorted, so nodes of one graph are contiguous. blockDim.x == F:
// thread f register-accumulates feature f over a strip of nodes, flushing
// one atomicAdd per graph-run boundary (up to 64x fewer atomics than per-element).
// Thread f==0 also accumulates run lengths into counts.
#define POOL_STRIP 64
__global__ void pool_strip_kernel(const float* __restrict__ h,
                                  const int* __restrict__ batch,
                                  float* __restrict__ sums,
                                  float* __restrict__ counts,
                                  int n_nodes, int F) {
  const int f  = threadIdx.x;                 // 0..F-1
  const int i0 = blockIdx.x * POOL_STRIP;
  if (i0 >= n_nodes) return;
  const int i1 = min(i0 + POOL_STRIP, n_nodes);

  int g = batch[i0];
  float acc = 0.0f;
  int run = 0;
  for (int i = i0; i < i1; ++i) {
    const int gi = batch[i];                  // uniform across block, cached
    if (gi != g) {
      atomicAdd(&sums[(size_t)g * F + f], acc);
      if (f == 0) atomicAdd(&counts[g], (float)run);
      g = gi;
      acc = 0.0f;
      run = 0;
    }
    acc += h[(size_t)i * F + f];
    ++run;
  }
  atomicAdd(&sums[(size_t)g * F + f], acc);
  if (f == 0) atomicAdd(&counts[g], (float)run);
}

// ---------------- head: mean -> FC -> log_softmax ----------------
// grid = n_graphs, block = n_classes (16). Each thread: one logit.
__global__ void head_kernel(const float* __restrict__ sums,
                            const float* __restrict__ counts,
                            const float* __restrict__ Wfc,
                            const float* __restrict__ bfc,
                            float* __restrict__ out,
                            int F, int C) {
  __shared__ float logit[64];
  int g = blockIdx.x;
  int c = threadIdx.x;
  float inv_cnt = 1.0f / fmaxf(counts[g], 1.0f);
  float acc = bfc[c];
  const float* srow = sums + (size_t)g * F;
  for (int k = 0; k < F; ++k)
    acc += (srow[k] * inv_cnt) * Wfc[(size_t)k * C + c];
  logit[c] = acc;
  __syncthreads();
  float mx = -INFINITY;
  for (int j = 0; j < C; ++j) mx = fmaxf(mx, logit[j]);
  float se = 0.0f;
  for (int j = 0; j < C; ++j) se += expf(logit[j] - mx);
  out[(size_t)g * C + c] = acc - mx - logf(se);
}

// ---------------------------------------------------------------------------

static inline int cdiv(long long a, int b) { return (int)((a + b - 1) / b); }

extern "C" void kernel_launch(void* const* d_in, const int* in_sizes, int n_in,
                              void* d_out, int out_size, void* d_ws, size_t ws_size,
                              hipStream_t stream) {
  const float* x      = (const float*)d_in[0];
  const int*   eidx   = (const int*)d_in[1];   // [2, E] flat
  const int*   batch  = (const int*)d_in[2];
  const float* W1     = (const float*)d_in[3];
  const float* b1     = (const float*)d_in[4];
  const float* W2     = (const float*)d_in[5];
  const float* b2     = (const float*)d_in[6];
  const float* Wfc    = (const float*)d_in[7];
  const float* bfc    = (const float*)d_in[8];
  float* out = (float*)d_out;

  const int NHID  = in_sizes[4];               // 128
  const int NCLS  = in_sizes[8];               // 16
  const int NFEAT = in_sizes[3] / NHID;        // 256
  const int NN    = in_sizes[0] / NFEAT;       // 50000
  const int E     = in_sizes[1] / 2;           // 600000
  const int NG    = out_size / NCLS;           // 64

  const int* src = eidx;
  const int* dst = eidx + E;

  // Workspace layout (byte offsets, packed back to back, no alignment padding).
  char* ws = (char*)d_ws;
  const size_t hw_bytes = (size_t)NN * NHID * sizeof(float);   // 25.6 MB
  float* hw    = (float*)(ws);                                  // GEMM output
  float* hbuf  = (float*)(ws + hw_bytes);                       // scatter target
  float* deg   = (float*)(ws + 2 * hw_bytes);
  float* dinv  = (float*)(ws + 2 * hw_bytes + (size_t)NN * sizeof(float));
  float* sums  = (float*)(ws + 2 * hw_bytes + (size_t)2 * NN * sizeof(float));
  float* cnts  = (float*)(ws + 2 * hw_bytes + (size_t)2 * NN * sizeof(float)
                             + (size_t)NG * NHID * sizeof(float));

  const int TPB = 256;
  const int wavesPerBlock = TPB / 32;          // 8 rows / edges per block

  // ---- degree & symmetric normalization ----
  deg_init_kernel<<<cdiv(NN, TPB), TPB, 0, stream>>>(deg, NN);
  deg_edge_kernel<<<cdiv(E, TPB), TPB, 0, stream>>>(dst, deg, E);
  dinv_kernel<<<cdiv(NN, TPB), TPB, 0, stream>>>(deg, dinv, NN);

  // ---- conv1: hw = x @ W1 ; h1 = scatter(hw) + selfloop + b1 ; ReLU ----
  {
    dim3 block(32 * (NHID / 16));              // 8 waves = 256 threads
    dim3 grid(NN / 16);                        // 3125 tiles of 16 rows
    gemm_wmma_f32_kernel<<<grid, block, 0, stream>>>(x, W1, hw, NN, NFEAT, NHID);
  }
  hipMemsetAsync(hbuf, 0, hw_bytes, stream);
  scatter_edge_kernel<<<cdiv(E, wavesPerBlock), TPB, 0, stream>>>(src, dst, dinv, hw, hbuf, E, NHID);
  bias_self_act_kernel<<<cdiv(NN, wavesPerBlock), TPB, 0, stream>>>(hbuf, hw, dinv, b1, NN, NHID, 1);

  // ---- conv2: hw = h1 @ W2 ; h2 = scatter(hw) + selfloop + b2 ----
  {
    dim3 block(32 * (NHID / 16));
    dim3 grid(NN / 16);
    gemm_wmma_f32_kernel<<<grid, block, 0, stream>>>(hbuf, W2, hw, NN, NHID, NHID);
  }
  // hbuf (h1) no longer needed after GEMM; reuse as conv2 scatter target.
  hipMemsetAsync(hbuf, 0, hw_bytes, stream);
  scatter_edge_kernel<<<cdiv(E, wavesPerBlock), TPB, 0, stream>>>(src, dst, dinv, hw, hbuf, E, NHID);
  bias_self_act_kernel<<<cdiv(NN, wavesPerBlock), TPB, 0, stream>>>(hbuf, hw, dinv, b2, NN, NHID, 0);

  // ---- global mean pool (sorted-run strips) ----
  hipMemsetAsync(sums, 0, (size_t)NG * NHID * sizeof(float), stream);
  hipMemsetAsync(cnts, 0, (size_t)NG * sizeof(float), stream);
  pool_strip_kernel<<<cdiv(NN, POOL_STRIP), NHID, 0, stream>>>(hbuf, batch, sums, cnts, NN, NHID);

  // ---- FC + log_softmax ----
  head_kernel<<<NG, NCLS, 0, stream>>>(sums, cnts, Wfc, bfc, out, NHID, NCLS);
}